// Retrace_24988119728619
// MI455X (gfx1250) — compile-verified
//
#include <hip/hip_runtime.h>
#include <cstdint>
#include <cstddef>

namespace {
constexpr float kGamma   = 0.99f;
constexpr float kEps     = 1e-10f;
constexpr int   kRows    = 4096;
constexpr int   kTp1     = 2049;
constexpr int   kT       = 2048;
constexpr int   kChunk   = 256;            // elements per chunk per row
constexpr int   kNChunk  = kT / kChunk;    // 8
constexpr int   kEPL     = kChunk / 32;    // 8 elements per lane
constexpr int   kWaves   = 4;              // waves (rows) per block
constexpr int   kStreams = 6;
}

// One wave32 per row. Backward linear-recurrence scan:
//   x[t] = a[t]*x[t+1] + d[t],  a[t] = GAMMA*min(tp[t+1]/max(bp[t+1],eps),1),
//   d[t] = r[t] + GAMMA*eQ[t+1] - tQ[t];  accumulate (x[t]-Q[t])^2.
// Data is staged into LDS (reversed-time order) with gfx1250 async loads
// (non-temporal: 201 MB streamed once > 192 MB L2), double-buffered one
// chunk ahead, synchronized via s_wait_asynccnt.
__global__ __launch_bounds__(kWaves * 32)
void retrace_scan_kernel(const float* __restrict__ Q,
                         const float* __restrict__ eQ,
                         const float* __restrict__ tQ,
                         const float* __restrict__ rw,
                         const float* __restrict__ tp,
                         const float* __restrict__ bp,
                         float* __restrict__ partials)
{
  __shared__ float lds[kWaves][2][kStreams][kChunk];
  const int lane = threadIdx.x & 31;
  const int w    = threadIdx.x >> 5;
  const int row  = blockIdx.x * kWaves + w;
  const size_t rb = (size_t)row * kTp1;

  // Streams needing index t+1 are staged with a +1 base shift.
  // Row stride 2049 floats => only 4B alignment guaranteed: B32 asyncs.
  const float* st[kStreams] = {
    Q + rb, eQ + rb + 1, tQ + rb, rw + rb, tp + rb + 1, bp + rb + 1
  };

  auto issue = [&](int c, int bf) {
    const int base = c * kChunk;
#pragma unroll
    for (int s = 0; s < kStreams; ++s) {
#pragma unroll
      for (int q = 0; q < kEPL; ++q) {
        const int j = q * 32 + lane;  // processing-order index within chunk
        const unsigned long long ga =
            (unsigned long long)(uintptr_t)(st[s] + (base + (kChunk - 1) - j));
        const unsigned loff = (unsigned)(uintptr_t)(&lds[w][bf][s][j]);
        asm volatile("global_load_async_to_lds_b32 %0, %1, off th:TH_LOAD_NT"
                     :: "v"(loff), "v"(ga) : "memory");
      }
    }
  };

  float acc    = 0.0f;  // sum of squared errors for this row
  float xcarry = 0.0f;  // running target value entering the chunk (x[t_end+1])

  issue(kNChunk - 1, (kNChunk - 1) & 1);

  for (int c = kNChunk - 1; c >= 0; --c) {
    if (c > 0) {
      issue(c - 1, (c - 1) & 1);                      // keep next chunk in flight
      asm volatile("s_wait_asynccnt 48" ::: "memory"); // current chunk resident
    } else {
      asm volatile("s_wait_asynccnt 0" ::: "memory");
    }
    const int bf = c & 1;

    float aa[kEPL], dd[kEPL], qq[kEPL];
#pragma unroll
    for (int h = 0; h < kEPL / 4; ++h) {
      const int j0 = kEPL * lane + 4 * h;   // 16B aligned in LDS
      float4 qv = *(const float4*)&lds[w][bf][0][j0];
      float4 ev = *(const float4*)&lds[w][bf][1][j0];
      float4 tv = *(const float4*)&lds[w][bf][2][j0];
      float4 rv = *(const float4*)&lds[w][bf][3][j0];
      float4 pv = *(const float4*)&lds[w][bf][4][j0];
      float4 bv = *(const float4*)&lds[w][bf][5][j0];
      const float qa[4] = {qv.x, qv.y, qv.z, qv.w};
      const float ea[4] = {ev.x, ev.y, ev.z, ev.w};
      const float ta[4] = {tv.x, tv.y, tv.z, tv.w};
      const float ra[4] = {rv.x, rv.y, rv.z, rv.w};
      const float pa[4] = {pv.x, pv.y, pv.z, pv.w};
      const float ba[4] = {bv.x, bv.y, bv.z, bv.w};
#pragma unroll
      for (int i = 0; i < 4; ++i) {
        const int k = 4 * h + i;
        const float cc = fminf(pa[i] / fmaxf(ba[i], kEps), 1.0f);
        aa[k] = kGamma * cc;
        dd[k] = ra[i] + kGamma * ea[i] - ta[i];
        qq[k] = qa[i];
      }
    }

    // Lane-local composition of 8 affine steps: x -> A*x + Bv
    float A = aa[0], Bv = dd[0];
#pragma unroll
    for (int k = 1; k < kEPL; ++k) {
      Bv = fmaf(aa[k], Bv, dd[k]);
      A  = aa[k] * A;
    }

    // Wave32 inclusive Hillis-Steele scan over affine pairs.
    float ai = A, bi = Bv;
#pragma unroll
    for (int off = 1; off < 32; off <<= 1) {
      const float pa_ = __shfl_up(ai, off, 32);
      const float pb_ = __shfl_up(bi, off, 32);
      if (lane >= off) { bi = fmaf(ai, pb_, bi); ai = ai * pa_; }
    }
    // Exclusive prefix for this lane.
    float ax = __shfl_up(ai, 1, 32);
    float bx = __shfl_up(bi, 1, 32);
    if (lane == 0) { ax = 1.0f; bx = 0.0f; }

    float x = fmaf(ax, xcarry, bx);
#pragma unroll
    for (int k = 0; k < kEPL; ++k) {
      x = fmaf(aa[k], x, dd[k]);      // x == target[t] for this element
      const float e = x - qq[k];
      acc = fmaf(e, e, acc);
    }

    // Carry = full-chunk aggregate applied to old carry (lane 31 inclusive).
    const float aT = __shfl(ai, 31, 32);
    const float bT = __shfl(bi, 31, 32);
    xcarry = fmaf(aT, xcarry, bT);
  }

  // Wave reduction of squared-error accumulator; one partial per row.
#pragma unroll
  for (int off = 16; off > 0; off >>= 1)
    acc += __shfl_xor(acc, off, 32);
  if (lane == 0) partials[row] = acc;
}

__global__ __launch_bounds__(256)
void retrace_reduce_kernel(const float* __restrict__ partials,
                           float* __restrict__ out)
{
  __shared__ float sh[256];
  float v = 0.0f;
  for (int i = threadIdx.x; i < kRows; i += 256) v += partials[i];
  sh[threadIdx.x] = v;
  __syncthreads();
  for (int s = 128; s > 0; s >>= 1) {
    if ((int)threadIdx.x < s) sh[threadIdx.x] += sh[threadIdx.x + s];
    __syncthreads();
  }
  if (threadIdx.x == 0)
    out[0] = sh[0] * (1.0f / ((float)kRows * (float)kT));
}

extern "C" void kernel_launch(void* const* d_in, const int* in_sizes, int n_in,
                              void* d_out, int out_size, void* d_ws, size_t ws_size,
                              hipStream_t stream)
{
  (void)in_sizes; (void)n_in; (void)out_size; (void)ws_size;
  const float* Q  = (const float*)d_in[0];
  const float* eQ = (const float*)d_in[1];
  const float* tQ = (const float*)d_in[2];
  const float* rw = (const float*)d_in[3];
  const float* tp = (const float*)d_in[4];
  const float* bp = (const float*)d_in[5];
  float* partials = (float*)d_ws;   // kRows floats = 16 KB scratch

  retrace_scan_kernel<<<kRows / kWaves, kWaves * 32, 0, stream>>>(
      Q, eQ, tQ, rw, tp, bp, partials);
  retrace_reduce_kernel<<<1, 256, 0, stream>>>(partials, (float*)d_out);
}